// MultiLevelEmbedding_34437047780006
// MI455X (gfx1250) — compile-verified
//
#include <hip/hip_runtime.h>

// MultiLevelEmbedding for MI455X (gfx1250).
// out[b,t,:] = emb_tables[level_ids[b,t], token_ids[b,t], :] + level_embed[level_ids[b,t], :]
//
// B=64, T=1024, D=512, L=4, VOCAB=258, all f32.
// Write-bandwidth-bound (134 MB streamed out; 2.1 MB L2-resident table).
// Strategy: wave-per-row, b128 vector memory ops, NT stores for the streaming
// output, and CDNA5 async global->LDS staging of the level-bias table.

typedef float v4f __attribute__((ext_vector_type(4)));

#define LVLS   4
#define VOCABSZ 258
#define DIM    512
#define D4     (DIM / 4)        // 128 float4 per row
#define TPB    256              // 8 waves per block

__global__ __launch_bounds__(TPB) void multilevel_embed_kernel(
    const int*   __restrict__ level_ids,   // [rows]
    const int*   __restrict__ token_ids,   // [rows]
    const float* __restrict__ emb_tables,  // [L, VOCAB, D]
    const float* __restrict__ level_embed, // [L, D]
    float*       __restrict__ out,         // [rows, D]
    int rows)
{
    // 8 KB bias table staged in LDS (320 KB/WGP available; this is cheap).
    __shared__ float sLevel[LVLS * DIM];

    // ---- Stage level_embed [L,D] into LDS via CDNA5 async global->LDS DMA.
    // 8192 bytes / (256 threads * 16 B) = 2 async b128 ops per thread.
    {
        const unsigned tid      = threadIdx.x;
        // Low 32 bits of a generic pointer to __shared__ are the LDS byte
        // address (shared-aperture tag lives in bits [63:32]).
        const unsigned lds_base = (unsigned)(unsigned long long)(&sLevel[0]);
#pragma unroll
        for (int i = 0; i < (LVLS * DIM * 4) / (TPB * 16); ++i) {
            const unsigned byte_off = (tid + (unsigned)i * TPB) * 16u;
            const unsigned lds_addr = lds_base + byte_off;
            // GVS mode: vdst = LDS addr VGPR, vaddr = 32-bit byte offset,
            // saddr = 64-bit SGPR base. Tracked on ASYNCcnt.
            asm volatile(
                "global_load_async_to_lds_b128 %0, %1, %2 offset:0"
                :
                : "v"(lds_addr), "v"(byte_off), "s"(level_embed)
                : "memory");
        }
        asm volatile("s_wait_asynccnt 0" ::: "memory");
    }
    __syncthreads();

    const int lane          = threadIdx.x & 31;
    const int waveInBlock   = threadIdx.x >> 5;
    const int wavesPerBlock = blockDim.x >> 5;
    const int numWaves      = gridDim.x * wavesPerBlock;
    int       wave          = blockIdx.x * wavesPerBlock + waveInBlock;

    const v4f* __restrict__ bias_all = (const v4f*)sLevel;

    // One wave32 per 512-float row: lane covers columns lane + 32*j (j=0..3),
    // i.e. 4x b128 loads + 4x b128 NT stores per row per lane. Row indices are
    // wave-uniform, so the two index loads coalesce to single cacheline hits.
    for (int row = wave; row < rows; row += numWaves) {
        const int lvl = level_ids[row];
        const int tok = token_ids[row];

        const v4f* __restrict__ src =
            (const v4f*)(emb_tables + ((size_t)lvl * VOCABSZ + (size_t)tok) * DIM);
        const v4f* __restrict__ bias = bias_all + (size_t)lvl * D4;
        v4f* __restrict__ dst        = (v4f*)(out + (size_t)row * DIM);

#pragma unroll
        for (int j = 0; j < 4; ++j) {
            const int c = lane + 32 * j;
            v4f v = src[c];         // global_load_b128 (L2/WGP$-resident table)
            v4f b = bias[c];        // ds_load_b128 from LDS
            v4f r = v + b;
            // Streaming 134 MB output: non-temporal so we don't evict the
            // 2 MB hot table from the near caches.
            __builtin_nontemporal_store(r, &dst[c]);
        }
    }
}

extern "C" void kernel_launch(void* const* d_in, const int* in_sizes, int n_in,
                              void* d_out, int out_size, void* d_ws, size_t ws_size,
                              hipStream_t stream)
{
    const int*   level_ids   = (const int*)d_in[0];   // [B*T] (int)
    const int*   token_ids   = (const int*)d_in[1];   // [B*T] (int)
    const float* emb_tables  = (const float*)d_in[2]; // [L, VOCAB, D]
    const float* level_embed = (const float*)d_in[3]; // [L, D]
    float*       out         = (float*)d_out;         // [B*T, D]

    const int rows = in_sizes[0];  // B*T = 65536

    // 2048 blocks x 256 threads = 16384 waves -> 4 rows/wave grid-stride.
    const int blocks = 2048;
    multilevel_embed_kernel<<<blocks, TPB, 0, stream>>>(
        level_ids, token_ids, emb_tables, level_embed, out, rows);
}